// dendriticNet_56994216018235
// MI455X (gfx1250) — compile-verified
//
#include <hip/hip_runtime.h>

#define DEVFN __device__ __forceinline__

#if defined(__has_builtin)
#if __has_builtin(__builtin_amdgcn_global_load_async_to_lds_b128) && \
    __has_builtin(__builtin_amdgcn_global_load_async_to_lds_b32)
#define HAS_ASYNC_LDS 1
#endif
#endif
#ifndef HAS_ASYNC_LDS
#define HAS_ASYNC_LDS 0
#endif

#define AS_GLOBAL __attribute__((address_space(1)))
#define AS_LDS __attribute__((address_space(3)))

typedef int v4i __attribute__((vector_size(4 * sizeof(int))));

#if HAS_ASYNC_LDS
#define WAIT_ASYNC(n) asm volatile("s_wait_asynccnt " #n ::: "memory")
#else
#define WAIT_ASYNC(n)
#endif
#define WAIT_DS0() asm volatile("s_wait_dscnt 0x0" ::: "memory")

namespace {
constexpr int   N    = 4096;
constexpr int   T    = 50;
constexpr float DT   = 0.1f;
constexpr float GA   = 0.8f;
constexpr float GB   = 1.0f;
constexpr float GD   = 1.0f;
constexpr float GLK  = 0.1f;
constexpr float GSOM = 0.8f;

constexpr int KC         = 32;                   // K chunk per stage
constexpr int LDS_STRIDE = KC + 4;               // 36 floats: 16B rows, bank-staggered
constexpr int WAVE_LDS   = 16 * LDS_STRIDE + KC; // 608 floats: A tile + x chunk
constexpr int DEPTH      = 3;                    // triple buffer (2 chunks in flight)
constexpr int WAVES      = 8;                    // 256 threads
constexpr int ROWS_PER_BLOCK = WAVES * 16;       // 128
constexpr int BLOCKS_PER_MAT = N / ROWS_PER_BLOCK; // 32
constexpr int KSPLIT     = 2;
constexpr int KHALF      = N / KSPLIT;           // 2048
constexpr int A_ITERS    = (16 * KC / 4) / 32;   // 4 float4 staging iters
}

typedef __attribute__((ext_vector_type(2))) float v2f;
typedef __attribute__((ext_vector_type(8))) float v8f;

DEVFN float softplus_f(float x) {
  return fmaxf(x, 0.0f) + log1pf(expf(-fabsf(x)));
}

struct MatvecArgs {
  const float* W[9];
  const float* X[9];
  float* out;  // 9 * KSPLIT * N partial results
};

// ---------------------------------------------------------------- init ------
__global__ __launch_bounds__(256) void dn_init_kernel(
    const float* __restrict__ data,
    const float* __restrict__ s0, const float* __restrict__ s1,
    const float* __restrict__ s2, const float* __restrict__ i0,
    const float* __restrict__ i1,
    float* __restrict__ s, float* __restrict__ rd) {
  const int idx = blockIdx.x * 256 + threadIdx.x;
  if (idx >= N) return;
  rd[idx] = softplus_f(data[idx]);
  s[0 * N + idx] = s0[idx];
  s[1 * N + idx] = s1[idx];
  s[2 * N + idx] = s2[idx];
  s[3 * N + idx] = i0[idx];
  s[4 * N + idx] = i1[idx];
}

// ---------------------------------------------------------------- prep ------
__global__ __launch_bounds__(256) void dn_prep_kernel(
    const float* __restrict__ s, float* __restrict__ rs,
    float* __restrict__ partials) {
  __shared__ float red1[256];
  __shared__ float red2[256];
  const int tid = threadIdx.x;
  const int idx = blockIdx.x * 256 + tid;  // grid = 16 -> idx < N always
  float r1 = 0.0f, r2 = 0.0f;
#pragma unroll
  for (int j = 0; j < 5; ++j) {
    const float v = softplus_f(s[j * N + idx]);
    rs[j * N + idx] = v;
    if (j == 1) r1 = v;
    if (j == 2) r2 = v;
  }
  red1[tid] = r1;
  red2[tid] = r2;
  __syncthreads();
  for (int off = 128; off > 0; off >>= 1) {
    if (tid < off) {
      red1[tid] += red1[tid + off];
      red2[tid] += red2[tid + off];
    }
    __syncthreads();
  }
  if (tid == 0) {
    partials[blockIdx.x]      = red1[0];
    partials[16 + blockIdx.x] = red2[0];
  }
}

// -------------------------------------------------------------- matvec ------
__global__ __launch_bounds__(256) void dn_matvec_kernel(MatvecArgs args) {
  __shared__ float lds[WAVES * DEPTH * WAVE_LDS];  // ~58 KB

  const int bid  = blockIdx.x;
  const int ks   = bid & (KSPLIT - 1);
  const int rb   = (bid >> 1) & (BLOCKS_PER_MAT - 1);
  const int mat  = bid >> 6;
  const int wave = threadIdx.x >> 5;
  const int lane = threadIdx.x & 31;
  const int j0   = rb * ROWS_PER_BLOCK + wave * 16;
  const int k0   = ks * KHALF;
  const int kend = k0 + KHALF;

  const float* __restrict__ W = args.W[mat];
  const float* __restrict__ X = args.X[mat];
  float* p0 = lds + wave * DEPTH * WAVE_LDS;  // wave-private triple buffer
  float* p1 = p0 + WAVE_LDS;
  float* p2 = p1 + WAVE_LDS;

  const int half = lane >> 4;  // 0 | 1
  const int l16  = lane & 15;

  // per-lane staging offsets, hoisted out of the stream loop
  unsigned wofs[A_ITERS];  // element offsets into W
  unsigned lofs[A_ITERS];  // element offsets into buf
#pragma unroll
  for (int it = 0; it < A_ITERS; ++it) {
    const int f = it * 32 + lane;  // float4 slot 0..127
    wofs[it] = (unsigned)((j0 + (f >> 3)) * N + (f & 7) * 4);
    lofs[it] = (unsigned)((f >> 3) * LDS_STRIDE + (f & 7) * 4);
  }

  auto stage = [&](float* buf, int kk) {
#if HAS_ASYNC_LDS
#pragma unroll
    for (int it = 0; it < A_ITERS; ++it)
      __builtin_amdgcn_global_load_async_to_lds_b128(
          (AS_GLOBAL v4i*)(W + wofs[it] + kk),
          (AS_LDS v4i*)(buf + lofs[it]), 0, 0);
    __builtin_amdgcn_global_load_async_to_lds_b32(
        (AS_GLOBAL int*)(X + kk + lane),
        (AS_LDS int*)(buf + 16 * LDS_STRIDE + lane), 0, 0);
#else
    float4 t[A_ITERS];
#pragma unroll
    for (int it = 0; it < A_ITERS; ++it)
      t[it] = *(const float4*)(W + wofs[it] + kk);
    float tx = X[kk + lane];
#pragma unroll
    for (int it = 0; it < A_ITERS; ++it)
      *(float4*)(buf + lofs[it]) = t[it];
    buf[16 * LDS_STRIDE + lane] = tx;
#endif
  };

  v8f c = {};
  auto compute = [&](const float* buf) {
    const float* ap = buf + l16 * LDS_STRIDE + 2 * half;
    const float* xp = buf + 16 * LDS_STRIDE + 2 * half;
#pragma unroll
    for (int k = 0; k < KC; k += 4) {
      const v2f a = *(const v2f*)(ap + k);
      const v2f b = *(const v2f*)(xp + k);
      c = __builtin_amdgcn_wmma_f32_16x16x4_f32(
          false, a, false, b, (short)0, c, false, false);
    }
  };

  // software pipeline: 2 chunks of DMA always in flight
  stage(p0, k0);
  stage(p1, k0 + KC);
  for (int kk = k0; kk < kend - 2 * KC; kk += KC) {  // 62 iterations
    WAIT_DS0();                 // buffer p2 reads (previous iter) finished
    stage(p2, kk + 2 * KC);
    WAIT_ASYNC(10);             // in-order: oldest chunk (p0) has landed
    __builtin_amdgcn_wave_barrier();
    compute(p0);
    __builtin_amdgcn_wave_barrier();
    float* t = p0; p0 = p1; p1 = p2; p2 = t;
  }
  WAIT_ASYNC(5);
  __builtin_amdgcn_wave_barrier();
  compute(p0);
  WAIT_ASYNC(0);
  __builtin_amdgcn_wave_barrier();
  compute(p1);

  // Every column of D is the matvec result; lane 0 holds M=0..7, lane 16 M=8..15.
  if ((lane & 15) == 0) {
    float* out = args.out + (size_t)(mat * KSPLIT + ks) * N + j0 + (half << 3);
    *(float4*)(out)     = make_float4(c[0], c[1], c[2], c[3]);
    *(float4*)(out + 4) = make_float4(c[4], c[5], c[6], c[7]);
  }
}

// -------------------------------------------------------------- update ------
__global__ __launch_bounds__(256) void dn_update_kernel(
    float* __restrict__ s, const float* __restrict__ mv,
    const float* __restrict__ partials) {
  const int idx = blockIdx.x * 256 + threadIdx.x;  // grid = 16
  float sum1 = 0.0f, sum2 = 0.0f;
#pragma unroll
  for (int b = 0; b < 16; ++b) {
    sum1 += partials[b];
    sum2 += partials[16 + b];
  }
  const float mean1 = sum1 * (1.0f / N);
  const float mean2 = sum2 * (1.0f / N);

  float r[9];
#pragma unroll
  for (int m = 0; m < 9; ++m)
    r[m] = mv[(size_t)(2 * m) * N + idx] + mv[(size_t)(2 * m + 1) * N + idx];

  const float s0v = s[0 * N + idx], s1v = s[1 * N + idx], s2v = s[2 * N + idx];
  const float i0v = s[3 * N + idx], i1v = s[4 * N + idx];

  const float vb0 = r[0], vi0 = r[1], va0 = r[2] + r[3];
  const float vb1 = r[4], vi1 = r[5], va1 = r[6] + r[7];
  const float vb2 = r[8];

  const float ds0 = -GLK * s0v + GB * (vb0 - s0v) + GA * (va0 - s0v);
  const float ds1 = -GLK * s1v + GB * (vb1 - s1v) + GA * (va1 - s1v);
  const float ds2 = -GLK * s2v + GB * (vb2 - s2v);
  const float di0 = -GLK * i0v + GD * (vi0 - i0v) + GSOM * (mean1 - i0v);
  const float di1 = -GLK * i1v + GD * (vi1 - i1v) + GSOM * (mean2 - i1v);

  s[0 * N + idx] = s0v + DT * ds0;
  s[1 * N + idx] = s1v + DT * ds1;
  s[2 * N + idx] = DT * ds2;  // faithful to reference: assignment, not +=
  s[3 * N + idx] = i0v + DT * di0;
  s[4 * N + idx] = i1v + DT * di1;
}

// ----------------------------------------------------------------- out ------
__global__ __launch_bounds__(256) void dn_out_kernel(
    const float* __restrict__ s, float* __restrict__ out) {
  const int idx = blockIdx.x * 256 + threadIdx.x;
  if (idx < 5 * N) out[idx] = s[idx];
}

// -------------------------------------------------------------- launch ------
extern "C" void kernel_launch(void* const* d_in, const int* in_sizes, int n_in,
                              void* d_out, int out_size, void* d_ws,
                              size_t ws_size, hipStream_t stream) {
  const float* data = (const float*)d_in[0];
  const float* s0i  = (const float*)d_in[1];
  const float* s1i  = (const float*)d_in[2];
  const float* s2i  = (const float*)d_in[3];
  const float* i0i  = (const float*)d_in[4];
  const float* i1i  = (const float*)d_in[5];
  const float* wpf0 = (const float*)d_in[6];
  const float* wpf1 = (const float*)d_in[7];
  const float* wpf2 = (const float*)d_in[8];
  const float* wpb0 = (const float*)d_in[9];
  const float* wpb1 = (const float*)d_in[10];
  const float* wip0 = (const float*)d_in[11];
  const float* wip1 = (const float*)d_in[12];
  const float* wpi0 = (const float*)d_in[13];
  const float* wpi1 = (const float*)d_in[14];

  float* ws   = (float*)d_ws;
  float* s    = ws;           // 5N  states
  float* rs   = ws + 5 * N;   // 5N  softplus(states)
  float* rd   = ws + 10 * N;  // N   softplus(data)
  float* mv   = ws + 11 * N;  // 18N partial matvecs
  float* part = ws + 29 * N;  // 32  mean partials

  dn_init_kernel<<<N / 256, 256, 0, stream>>>(data, s0i, s1i, s2i, i0i, i1i,
                                              s, rd);

  MatvecArgs args;
  args.W[0] = wpf0; args.X[0] = rd;
  args.W[1] = wip0; args.X[1] = rs + 0 * N;
  args.W[2] = wpi0; args.X[2] = rs + 3 * N;
  args.W[3] = wpb0; args.X[3] = rs + 1 * N;
  args.W[4] = wpf1; args.X[4] = rs + 0 * N;
  args.W[5] = wip1; args.X[5] = rs + 1 * N;
  args.W[6] = wpi1; args.X[6] = rs + 4 * N;
  args.W[7] = wpb1; args.X[7] = rs + 2 * N;
  args.W[8] = wpf2; args.X[8] = rs + 1 * N;
  args.out = mv;

  const int mv_grid = 9 * BLOCKS_PER_MAT * KSPLIT;  // 576 blocks
  for (int t = 0; t < T; ++t) {
    dn_prep_kernel<<<16, 256, 0, stream>>>(s, rs, part);
    dn_matvec_kernel<<<mv_grid, 256, 0, stream>>>(args);
    dn_update_kernel<<<16, 256, 0, stream>>>(s, mv, part);
  }
  dn_out_kernel<<<(5 * N + 255) / 256, 256, 0, stream>>>(s, (float*)d_out);
}